// SinkhornDistance_44744969290142
// MI455X (gfx1250) — compile-verified
//
#include <hip/hip_runtime.h>
#include <hip/hip_bf16.h>
#include <math.h>

#define EPS      0.1f
#define INV_EPS  10.0f
#define TINY     1e-16f
#define NORM_EPS 1e-8f
#define NB  32
#define N   512
#define D   256
#define MAT (N * N)        // 262144 per batch

typedef float v2f __attribute__((ext_vector_type(2)));
typedef float v8f __attribute__((ext_vector_type(8)));

// ---------------------------------------------------------------------------
// Row L2 norms: one wave (32 lanes) per 256-element row. 16384 rows total.
// ---------------------------------------------------------------------------
__global__ __launch_bounds__(256)
void row_norms_kernel(const float* __restrict__ X, float* __restrict__ W) {
    const int row  = blockIdx.x * 8 + (threadIdx.x >> 5);
    const int lane = threadIdx.x & 31;
    const float* p = X + (size_t)row * D;
    float s = 0.f;
    #pragma unroll
    for (int d = 0; d < D / 32; ++d) {
        float t = p[lane + d * 32];
        s += t * t;
    }
    #pragma unroll
    for (int off = 16; off > 0; off >>= 1) s += __shfl_xor(s, off, 32);
    if (lane == 0) W[row] = sqrtf(s);
}

// ---------------------------------------------------------------------------
// Init u=1, v=1, alpha=0, beta=0, cost=0
// ---------------------------------------------------------------------------
__global__ void init_state_kernel(float* u, float* v, float* alpha, float* beta,
                                  float* cost) {
    const int i = blockIdx.x * 256 + threadIdx.x;
    if (i < NB * N) {
        u[i] = 1.f;  v[i] = 1.f;  alpha[i] = 0.f;  beta[i] = 0.f;
    }
    if (i < NB) cost[i] = 0.f;
}

// ---------------------------------------------------------------------------
// Cosine-distance batched GEMM via V_WMMA_F32_16X16X4_F32.
// grid = (32 batches, 4 n-groups, 8 m-strips); block = 256 (8 waves).
// Each wave computes a 16(n) x 64(m) strip, reusing its A fragment across
// four B tiles. K = 256 consumed 4 at a time (64 WMMAs per tile).
// A layout: lane l holds row l%16, K = 2*(l/16)+{0,1}  -> float2 load.
// B layout: lane l holds col l%16, K = 2*(l/16)+{0,1}  -> float2 load.
// ---------------------------------------------------------------------------
__global__ __launch_bounds__(256)
void cosine_gemm_kernel(const float* __restrict__ x, const float* __restrict__ y,
                        const float* __restrict__ w1, const float* __restrict__ w2,
                        float* __restrict__ C) {
    const int b    = blockIdx.x;
    const int wave = threadIdx.x >> 5;
    const int lane = threadIdx.x & 31;
    const int half = lane >> 4;      // 0 or 1
    const int lrow = lane & 15;
    const int n0   = blockIdx.y * 128 + wave * 16;
    const int m0   = blockIdx.z * 64;

    const float* xb = x + ((size_t)b * N + n0 + lrow) * D + half * 2;
    const float* yb = y + ((size_t)b * N + m0 + lrow) * D + half * 2;

    v8f acc[4] = {};
    #pragma unroll 4
    for (int k0 = 0; k0 < D; k0 += 4) {
        v2f a = *(const v2f*)(xb + k0);
        #pragma unroll
        for (int j = 0; j < 4; ++j) {
            v2f bf = *(const v2f*)(yb + (size_t)j * 16 * D + k0);
            acc[j] = __builtin_amdgcn_wmma_f32_16x16x4_f32(
                false, a, false, bf, (short)0, acc[j], false, false);
        }
    }

    // cosine transform + store.  VGPR r of a tile: row = r + 8*half, col = lrow.
    float w1r[8];
    #pragma unroll
    for (int r = 0; r < 8; ++r) w1r[r] = w1[b * N + n0 + 8 * half + r];

    float* Cb = C + (size_t)b * MAT;
    #pragma unroll
    for (int j = 0; j < 4; ++j) {
        const int   mcol = m0 + j * 16 + lrow;
        const float w2c  = w2[b * N + mcol];
        #pragma unroll
        for (int r = 0; r < 8; ++r) {
            const int   nrow  = n0 + 8 * half + r;
            const float denom = fmaxf(w1r[r] * w2c, NORM_EPS);
            Cb[(size_t)nrow * N + mcol] = 1.f - acc[j][r] / denom;
        }
    }
}

// ---------------------------------------------------------------------------
// Row update:  u[b,n] = (1/N) / (sum_m exp((-C+a_n+b_m)/eps) * v[m] + tiny)
// grid = (64 row-groups, 32 batches); block = 256 (8 waves, one row each).
// ---------------------------------------------------------------------------
__global__ __launch_bounds__(256)
void u_update_kernel(const float* __restrict__ C, const float* __restrict__ v,
                     const float* __restrict__ beta, const float* __restrict__ alpha,
                     float* __restrict__ u) {
    const int b = blockIdx.y;
    __shared__ float sv[N];
    __shared__ float sb[N];
    for (int t = threadIdx.x; t < N; t += 256) {
        sv[t] = v[b * N + t];
        sb[t] = beta[b * N + t];
    }
    __syncthreads();

    const int wave = threadIdx.x >> 5;
    const int lane = threadIdx.x & 31;
    const int n    = blockIdx.x * 8 + wave;
    const float al = alpha[b * N + n];
    const float* Crow = C + ((size_t)b * N + n) * N;

    float s = 0.f;
    #pragma unroll 4
    for (int m = lane; m < N; m += 32)
        s += expf((-Crow[m] + al + sb[m]) * INV_EPS) * sv[m];
    #pragma unroll
    for (int off = 16; off > 0; off >>= 1) s += __shfl_xor(s, off, 32);
    if (lane == 0) u[b * N + n] = (1.f / N) / (s + TINY);
}

// ---------------------------------------------------------------------------
// Column update: v[b,m] = (1/N)/(sum_n K^T u + tiny); beta updated in place
// (each thread owns exactly one beta[b,m]).  grid = (32 batches, 4 col-groups),
// block = 128; column reads are coalesced across the block.
// ---------------------------------------------------------------------------
__global__ __launch_bounds__(128)
void v_update_kernel(const float* __restrict__ C, const float* __restrict__ u,
                     float* __restrict__ v, const float* __restrict__ alpha,
                     float* __restrict__ beta) {
    const int b = blockIdx.x;
    __shared__ float sa[N];
    __shared__ float su[N];
    for (int t = threadIdx.x; t < N; t += 128) {
        sa[t] = alpha[b * N + t];
        su[t] = u[b * N + t];
    }
    __syncthreads();

    const int m  = blockIdx.y * 128 + threadIdx.x;
    const float be = beta[b * N + m];
    const float* Cb = C + (size_t)b * MAT;

    float s = 0.f;
    #pragma unroll 4
    for (int n = 0; n < N; ++n)
        s += expf((-Cb[(size_t)n * N + m] + sa[n] + be) * INV_EPS) * su[n];

    const float vv = (1.f / N) / (s + TINY);
    v[b * N + m]    = vv;
    beta[b * N + m] = be + EPS * logf(vv);
}

// alpha <- alpha + eps*log(u)   (separate kernel: avoids RAW race vs v_update)
__global__ void alpha_update_kernel(float* __restrict__ alpha,
                                    const float* __restrict__ u) {
    const int i = blockIdx.x * 256 + threadIdx.x;
    alpha[i] += EPS * logf(u[i]);
}

// ---------------------------------------------------------------------------
// pi = exp(log u + (-C+alpha+beta)/eps + log v);  cost[b] = sum(pi*C)
// grid = (32 batches, 64 chunks of 4096 elems); block = 256.
// ---------------------------------------------------------------------------
__global__ __launch_bounds__(256)
void finalize_kernel(const float* __restrict__ C, const float* __restrict__ u,
                     const float* __restrict__ v, const float* __restrict__ alpha,
                     const float* __restrict__ beta, float* __restrict__ pi,
                     float* __restrict__ cost) {
    const int b = blockIdx.x;
    const float* Cb  = C  + (size_t)b * MAT;
    float*       pib = pi + (size_t)b * MAT;
    const int base = blockIdx.y * 4096;

    float local = 0.f;
    #pragma unroll 2
    for (int k = 0; k < 16; ++k) {
        const int idx = base + k * 256 + threadIdx.x;
        const int n = idx >> 9;
        const int m = idx & (N - 1);
        const float Cv = Cb[idx];
        const float e  = logf(u[b * N + n]) + logf(v[b * N + m]) +
                         (-Cv + alpha[b * N + n] + beta[b * N + m]) * INV_EPS;
        const float p = expf(e);
        pib[idx] = p;
        local += p * Cv;
    }

    __shared__ float red[256];
    red[threadIdx.x] = local;
    __syncthreads();
    #pragma unroll
    for (int s = 128; s > 0; s >>= 1) {
        if (threadIdx.x < s) red[threadIdx.x] += red[threadIdx.x + s];
        __syncthreads();
    }
    if (threadIdx.x == 0) atomicAdd(&cost[b], red[0]);
}

// ---------------------------------------------------------------------------
extern "C" void kernel_launch(void* const* d_in, const int* in_sizes, int n_in,
                              void* d_out, int out_size, void* d_ws, size_t ws_size,
                              hipStream_t stream) {
    const float* x = (const float*)d_in[0];   // [32,512,256]
    const float* y = (const float*)d_in[1];   // [32,512,256]

    float* out  = (float*)d_out;
    float* cost = out;                        // [32]
    float* pi   = out + NB;                   // [32,512,512]
    float* Cm   = out + NB + (size_t)NB * MAT;// [32,512,512]

    float* ws    = (float*)d_ws;
    float* u     = ws;
    float* v     = ws + 1 * NB * N;
    float* alpha = ws + 2 * NB * N;
    float* beta  = ws + 3 * NB * N;
    float* w1    = ws + 4 * NB * N;
    float* w2    = ws + 5 * NB * N;

    // norms + state init
    row_norms_kernel<<<NB * N / 8, 256, 0, stream>>>(x, w1);
    row_norms_kernel<<<NB * N / 8, 256, 0, stream>>>(y, w2);
    init_state_kernel<<<(NB * N + 255) / 256, 256, 0, stream>>>(u, v, alpha, beta, cost);

    // cosine-distance GEMM (WMMA f32)
    cosine_gemm_kernel<<<dim3(NB, 4, 8), 256, 0, stream>>>(x, y, w1, w2, Cm);

    // 15 Sinkhorn iterations
    for (int it = 0; it < 15; ++it) {
        u_update_kernel<<<dim3(N / 8, NB), 256, 0, stream>>>(Cm, v, beta, alpha, u);
        v_update_kernel<<<dim3(NB, 4), 128, 0, stream>>>(Cm, u, v, alpha, beta);
        alpha_update_kernel<<<NB * N / 256, 256, 0, stream>>>(alpha, u);
    }

    // transport plan + per-batch cost
    finalize_kernel<<<dim3(NB, 64), 256, 0, stream>>>(Cm, u, v, alpha, beta, pi, cost);
}